// LSTMPredictor_44435731644910
// MI455X (gfx1250) — compile-verified
//
#include <hip/hip_runtime.h>
#include <stdint.h>

// ---------------------------------------------------------------------------
// LSTM predictor for MI455X (gfx1250, wave32, WMMA).
//
// Decomposition:
//   K1 (WMMA GEMM) : x_proj[T,712] = feats[T,80] @ W_ih^T + (b_ih + b_hh)
//   K2 (scale)     : scale[t] = max(||feats[t]||, 1e-6)
//   K3 (persistent): sequential LSTM recurrence, W_hh as f16 in LDS (256KB),
//                    f32 accumulation; logs h history HB[t+1] = h_t, HB[0]=0,
//                    row stride padded to 180 (pad cols zeroed).
//   K4 (WMMA GEMM) : preds[t,80] = tanh(HB[t] @ C^T) * scale[t]
//
// Workspace (floats): x_proj T*712  |  HB (T+1)*180  |  scale T   (~468 MB)
// ---------------------------------------------------------------------------

#define T_STEPS 131072
#define HD      178
#define HP      180     // padded hidden (stride for HB / LDS h)
#define G4      712     // 4*HD gate rows
#define NI      80      // N_MELS

typedef float v2f __attribute__((ext_vector_type(2)));
typedef float v8f __attribute__((ext_vector_type(8)));

// ---------------------------------------------------------------------------
// K1: x_proj = feats @ W_ih^T + bias, via V_WMMA_F32_16X16X4_F32.
// One wave per 16x16 output tile. N padded 712 -> 720 (45 tiles), stores guarded.
// A lanes: rows M = lane%16, K = kk + 2*(lane/16) .. +1   (f32 A layout)
// B lanes: cols N = lane%16, K = kk + 2*(lane/16) .. +1   (B[k][n] = W_ih[n][k])
// ---------------------------------------------------------------------------
__global__ __launch_bounds__(256) void k1_xproj(const float* __restrict__ feats,
                                                const float* __restrict__ Wih,
                                                const float* __restrict__ bih,
                                                const float* __restrict__ bhh,
                                                float* __restrict__ xproj,
                                                unsigned numNT, unsigned numTiles) {
    unsigned wave = blockIdx.x * (blockDim.x >> 5) + (threadIdx.x >> 5);
    if (wave >= numTiles) return;                 // wave-uniform exit (EXEC stays full)
    unsigned lane = threadIdx.x & 31u;
    unsigned hi = lane >> 4, lm = lane & 15u;
    unsigned mt = wave / numNT, nt = wave % numNT;
    unsigned m0 = mt * 16u, n0 = nt * 16u;

    const float* arow = feats + (size_t)(m0 + lm) * NI + 2u * hi;
    unsigned n  = n0 + lm;
    unsigned nc = (n < (unsigned)G4) ? n : (unsigned)(G4 - 1);   // clamp OOB load rows
    const float* brow = Wih + (size_t)nc * NI + 2u * hi;

    v8f acc = {0.f, 0.f, 0.f, 0.f, 0.f, 0.f, 0.f, 0.f};
    #pragma unroll
    for (int k = 0; k < NI; k += 4) {
        v2f a = *(const v2f*)(arow + k);
        v2f b = *(const v2f*)(brow + k);
        acc = __builtin_amdgcn_wmma_f32_16x16x4_f32(false, a, false, b,
                                                    (short)0, acc, false, false);
    }
    if (n < (unsigned)G4) {
        float bs = bih[n] + bhh[n];
        float* out = xproj + (size_t)(m0 + 8u * hi) * G4 + n;
        #pragma unroll
        for (int r = 0; r < 8; ++r)
            out[(size_t)r * G4] = acc[r] + bs;    // row m0 + r + 8*hi, col n
    }
}

// ---------------------------------------------------------------------------
// K2: per-row L2 norm of feats.
// ---------------------------------------------------------------------------
__global__ __launch_bounds__(256) void k2_scale(const float* __restrict__ feats,
                                                float* __restrict__ scale) {
    unsigned t = blockIdx.x * blockDim.x + threadIdx.x;
    if (t >= (unsigned)T_STEPS) return;
    const float4* f = (const float4*)(feats + (size_t)t * NI);
    float s = 0.f;
    #pragma unroll
    for (int i = 0; i < NI / 4; ++i) {
        float4 v = f[i];
        s += v.x * v.x + v.y * v.y + v.z * v.z + v.w * v.w;
    }
    s = sqrtf(s);
    scale[t] = (s > 1e-6f) ? s : 1e-6f;
}

// ---------------------------------------------------------------------------
// K3: persistent sequential LSTM. Single workgroup, 768 threads (24 waves).
// Dynamic LDS layout (bytes):
//   [0,              712*90*4)  W_hh as packed f16 pairs, row stride 90 dwords
//   [+0,             HP*4)      h (f32, pads [178..179] = 0)
//   [+HP*4,          G4*4)      gates (f32)
// ---------------------------------------------------------------------------
#define K3_WBYTES (G4 * 90 * 4)
#define K3_SMEM   (K3_WBYTES + HP * 4 + G4 * 4)

extern __shared__ char smemB[];

__global__ __launch_bounds__(768, 1) void k3_recur(const float* __restrict__ Whh,
                                                   const float* __restrict__ xproj,
                                                   float* __restrict__ HB) {
    unsigned tid = threadIdx.x;
    uint32_t* wl = (uint32_t*)smemB;
    float*    hl = (float*)(smemB + K3_WBYTES);
    float*    gl = (float*)(smemB + K3_WBYTES + HP * 4);

    // One-time: load W_hh, convert to f16 pairs, pad cols 178..179 with 0.
    for (unsigned idx = tid; idx < (unsigned)(G4 * 90); idx += blockDim.x) {
        unsigned r = idx / 90u, p = idx % 90u, j0 = 2u * p;
        float w0 = (j0     < (unsigned)HD) ? Whh[(size_t)r * HD + j0]      : 0.f;
        float w1 = (j0 + 1 < (unsigned)HD) ? Whh[(size_t)r * HD + j0 + 1u] : 0.f;
        union { _Float16 h[2]; uint32_t u; } pk;
        pk.h[0] = (_Float16)w0;
        pk.h[1] = (_Float16)w1;
        wl[idx] = pk.u;
    }
    if (tid < (unsigned)HP) {
        hl[tid] = 0.f;        // h_{-1} = 0 (+ pads)
        HB[tid] = 0.f;        // HB row 0 = h_{-1}
    }
    float c = 0.f;
    __syncthreads();

    float xp_cur = (tid < (unsigned)G4) ? xproj[tid] : 0.f;

    for (int t = 0; t < T_STEPS; ++t) {
        // Prefetch next step's x_proj row element (hidden under the dot).
        int tn = (t + 1 < T_STEPS) ? (t + 1) : t;
        float xp_next = (tid < (unsigned)G4) ? xproj[(size_t)tn * G4 + tid] : 0.f;

        if (tid < (unsigned)G4) {
            const uint32_t* wr = wl + (size_t)tid * 90u;
            float acc = 0.f;
            #pragma unroll 3
            for (int p = 0; p < 45; ++p) {
                float4 h4 = ((const float4*)hl)[p];     // broadcast (same addr all lanes)
                uint2  wp = ((const uint2*)wr)[p];      // b64, conflict-free banks
                union { uint32_t u; _Float16 h[2]; } a, b;
                a.u = wp.x; b.u = wp.y;
                acc += (float)a.h[0] * h4.x;
                acc += (float)a.h[1] * h4.y;
                acc += (float)b.h[0] * h4.z;
                acc += (float)b.h[1] * h4.w;
            }
            gl[tid] = xp_cur + acc;
        }
        __syncthreads();

        if (tid < (unsigned)HD) {
            float gi = gl[tid];
            float gf = gl[HD + tid];
            float gg = gl[2 * HD + tid];
            float go = gl[3 * HD + tid];
            float i_g = 1.f / (1.f + __expf(-gi));
            float f_g = 1.f / (1.f + __expf(-gf));
            float ag  = fabsf(gg);
            float eg  = __expf(-2.f * ag);
            float tg  = (1.f - eg) / (1.f + eg);
            float g_g = (gg < 0.f) ? -tg : tg;
            float o_g = 1.f / (1.f + __expf(-go));
            c = f_g * c + i_g * g_g;
            float ac  = fabsf(c);
            float ec  = __expf(-2.f * ac);
            float tc  = (1.f - ec) / (1.f + ec);
            float thc = (c < 0.f) ? -tc : tc;
            float hn  = o_g * thc;
            hl[tid] = hn;
            HB[(size_t)(t + 1) * HP + tid] = hn;     // h_t at row t+1
        } else if (tid < (unsigned)HP) {
            HB[(size_t)(t + 1) * HP + tid] = 0.f;    // keep pad cols zero
        }
        __syncthreads();
        xp_cur = xp_next;
    }
}

// ---------------------------------------------------------------------------
// K4: preds[t] = tanh(HB[t] @ C^T) * scale[t], via f32 WMMA.
// K padded 178 -> 180: HB pad cols are zero; C^T tail (k=178,179, only the
// hi-lane half of the last iteration) masked to zero.
// N = 80 = 5 tiles exactly.
// ---------------------------------------------------------------------------
__global__ __launch_bounds__(256) void k4_pred(const float* __restrict__ HB,
                                               const float* __restrict__ Cm,
                                               const float* __restrict__ scale,
                                               float* __restrict__ preds,
                                               unsigned numTiles) {
    unsigned wave = blockIdx.x * (blockDim.x >> 5) + (threadIdx.x >> 5);
    if (wave >= numTiles) return;
    unsigned lane = threadIdx.x & 31u;
    unsigned hi = lane >> 4, lm = lane & 15u;
    unsigned mt = wave / 5u, nt = wave % 5u;
    unsigned m0 = mt * 16u, n0 = nt * 16u;

    const float* arow = HB + (size_t)(m0 + lm) * HP + 2u * hi;
    const float* brow = Cm + (size_t)(n0 + lm) * HD + 2u * hi;

    v8f acc = {0.f, 0.f, 0.f, 0.f, 0.f, 0.f, 0.f, 0.f};
    #pragma unroll 5
    for (int k = 0; k < HP; k += 4) {
        v2f a = *(const v2f*)(arow + k);       // pads are zeroed in HB
        v2f b;
        if ((unsigned)(k + 2u * hi) < (unsigned)HD) {
            b = *(const v2f*)(brow + k);
        } else {
            b.x = 0.f; b.y = 0.f;              // C has no k=178,179
        }
        acc = __builtin_amdgcn_wmma_f32_16x16x4_f32(false, a, false, b,
                                                    (short)0, acc, false, false);
    }
    unsigned col = n0 + lm;                    // always < 80
    #pragma unroll
    for (int r = 0; r < 8; ++r) {
        unsigned row = m0 + (unsigned)r + 8u * hi;
        preds[(size_t)row * NI + col] = tanhf(acc[r]) * scale[row];
    }
}

// ---------------------------------------------------------------------------
extern "C" void kernel_launch(void* const* d_in, const int* in_sizes, int n_in,
                              void* d_out, int out_size, void* d_ws, size_t ws_size,
                              hipStream_t stream) {
    const float* feats = (const float*)d_in[0];
    const float* Wih   = (const float*)d_in[1];
    const float* Whh   = (const float*)d_in[2];
    const float* bih   = (const float*)d_in[3];
    const float* bhh   = (const float*)d_in[4];
    const float* Cm    = (const float*)d_in[5];
    float* preds = (float*)d_out;

    float* ws    = (float*)d_ws;
    float* xproj = ws;                                       // T*712
    float* HB    = xproj + (size_t)T_STEPS * G4;             // (T+1)*180
    float* scl   = HB + (size_t)(T_STEPS + 1) * HP;          // T

    // K1: 8192 M-tiles x 45 N-tiles, 8 waves (tiles) per 256-thread block.
    {
        unsigned numNT = 45, numTiles = (T_STEPS / 16) * numNT;   // 368640
        unsigned blocks = (numTiles + 7) / 8;                     // 46080
        k1_xproj<<<blocks, 256, 0, stream>>>(feats, Wih, bih, bhh, xproj,
                                             numNT, numTiles);
    }
    // K2: scale.
    k2_scale<<<T_STEPS / 256, 256, 0, stream>>>(feats, scl);

    // K3: persistent recurrence, 1 workgroup, ~254 KB dynamic LDS.
    k3_recur<<<1, 768, K3_SMEM, stream>>>(Whh, xproj, HB);

    // K4: 8192 M-tiles x 5 N-tiles.
    {
        unsigned numTiles = (T_STEPS / 16) * 5;                   // 40960
        unsigned blocks = (numTiles + 7) / 8;                     // 5120
        k4_pred<<<blocks, 256, 0, stream>>>(HB, Cm, scl, preds, numTiles);
    }
}